// Maxish_25460566130920
// MI455X (gfx1250) — compile-verified
//
#include <hip/hip_runtime.h>
#include <stdint.h>

// Maxish: sign-conditional softmax-weighted max reduce over last dim.
// x: [B,T,X,N] f32 with N=256 -> out [B,T,X] f32.
// Bandwidth-bound (512 MiB @ 23.3 TB/s ~= 23 us floor). Strategy:
//  - per-wave double-buffered async global->LDS staging (CDNA5 async engine:
//    global_load_async_to_lds_b128 + s_wait_asynccnt)
//  - wave32 ds_swizzle butterfly reductions (no cross-wave sync needed)
//  - algebraic simplification: pos = m * sum(exp((s+1)rt)) / sum(exp(s*rt)),
//    neg = m exactly (softmax weights sum to 1).
//  - exp args folded to one FMA each: exp2(v*k - m*k), c = -m*k precomputed.

#define SMALL_NUMBER 1e-8f
#define LOG2E 1.4426950408889634f

#define WAVES 8    // 256 threads / 32
#define ROW_N 256  // floats per reduced row
#define ROW_B 1024 // bytes per row
#define ITERS 16   // rows per wave

#if __has_builtin(__builtin_amdgcn_exp2f)
#define EXP2F(x) __builtin_amdgcn_exp2f(x)
#else
#define EXP2F(x) exp2f(x)  // HIP device overload -> __ocml_exp2_f32 -> v_exp_f32
#endif

#if __has_builtin(__builtin_amdgcn_rcpf)
#define RCPF(x) __builtin_amdgcn_rcpf(x)
#else
#define RCPF(x) (1.0f / (x))
#endif

// wave32 butterfly lane shuffles via LDS swizzle unit (no LDS memory traffic)
#define SWZF(v, pat) \
  __int_as_float(__builtin_amdgcn_ds_swizzle(__float_as_int(v), (pat)))

__device__ __forceinline__ float wave_red_max(float v) {
  v = fmaxf(v, SWZF(v, 0x401F)); // xor 16
  v = fmaxf(v, SWZF(v, 0x201F)); // xor 8
  v = fmaxf(v, SWZF(v, 0x101F)); // xor 4
  v = fmaxf(v, SWZF(v, 0x081F)); // xor 2
  v = fmaxf(v, SWZF(v, 0x041F)); // xor 1
  return v;
}

__device__ __forceinline__ float wave_red_sum(float v) {
  v += SWZF(v, 0x401F);
  v += SWZF(v, 0x201F);
  v += SWZF(v, 0x101F);
  v += SWZF(v, 0x081F);
  v += SWZF(v, 0x041F);
  return v;
}

__global__ __launch_bounds__(256) void Maxish_25460566130920_kernel(
    const float* __restrict__ x, const float* __restrict__ scale_p,
    float* __restrict__ out, unsigned nrows) {
  // 2 pipeline buffers x 8 waves x 1KiB row = 16 KiB LDS
  __shared__ __align__(16) float stage[2][WAVES][ROW_N];

  const unsigned lane = threadIdx.x & 31u;
  const unsigned wave = threadIdx.x >> 5;
  const float scale = *scale_p;

  const unsigned stride = gridDim.x * WAVES;     // rows advanced per step
  const unsigned row0 = blockIdx.x * WAVES + wave;

  // LDS byte address for this lane's 16B slice of this wave's row slot.
  // (low 32 bits of the flat pointer are the LDS offset on gfx1250)
  const uint32_t lds_base =
      (uint32_t)(uintptr_t)(&stage[0][wave][0]) + lane * 16u;
  const uint64_t xbase = (uint64_t)(uintptr_t)x;

  // Async DMA one 1KiB row into this wave's LDS slot (2 x B128 per lane;
  // inst offset moves BOTH the global and LDS sides).
  auto issue = [&](unsigned buf, unsigned r) {
    uint32_t voff = (r << 10) + lane * 16u;           // byte offset into x
    uint32_t loff = lds_base + buf * (WAVES * ROW_B); // LDS byte address
    asm volatile(
        "global_load_async_to_lds_b128 %0, %1, %2\n\t"
        "global_load_async_to_lds_b128 %0, %1, %2 offset:512"
        :
        : "v"(loff), "v"(voff), "s"(xbase)
        : "memory");
  };

  if (row0 < nrows) issue(0u, row0);

  for (unsigned i = 0; i < ITERS; ++i) {
    const unsigned cur = row0 + i * stride;
    const unsigned nxt = cur + stride;

    // Prefetch next row into the other buffer, then wait only for the
    // current buffer (ASYNCcnt completes in order => wait <=2 drains buf i).
    if ((i + 1u < ITERS) && (nxt < nrows)) {
      issue((i + 1u) & 1u, nxt);
      asm volatile("s_wait_asynccnt 0x2" ::: "memory");
    } else {
      asm volatile("s_wait_asynccnt 0x0" ::: "memory");
    }
    if (cur >= nrows) continue;

    const float4* p =
        reinterpret_cast<const float4*>(&stage[i & 1u][wave][lane * 8u]);
    float4 a = p[0];
    float4 b = p[1];

    float m = fmaxf(fmaxf(fmaxf(a.x, a.y), fmaxf(a.z, a.w)),
                    fmaxf(fmaxf(b.x, b.y), fmaxf(b.z, b.w)));
    m = wave_red_max(m);

    const float inv = RCPF(m + SMALL_NUMBER);
    const float k0 = LOG2E * inv;
    const float k1 = scale * k0;          // exp(s*rt)     = exp2(v*k1 + c1)
    const float k2 = (scale + 1.0f) * k0; // exp((s+1)*rt) = exp2(v*k2 + c2)
    const float c1 = -m * k1;
    const float c2 = -m * k2;

    float s1 = 0.0f, s2 = 0.0f;
    s1 += EXP2F(fmaf(a.x, k1, c1)); s2 += EXP2F(fmaf(a.x, k2, c2));
    s1 += EXP2F(fmaf(a.y, k1, c1)); s2 += EXP2F(fmaf(a.y, k2, c2));
    s1 += EXP2F(fmaf(a.z, k1, c1)); s2 += EXP2F(fmaf(a.z, k2, c2));
    s1 += EXP2F(fmaf(a.w, k1, c1)); s2 += EXP2F(fmaf(a.w, k2, c2));
    s1 += EXP2F(fmaf(b.x, k1, c1)); s2 += EXP2F(fmaf(b.x, k2, c2));
    s1 += EXP2F(fmaf(b.y, k1, c1)); s2 += EXP2F(fmaf(b.y, k2, c2));
    s1 += EXP2F(fmaf(b.z, k1, c1)); s2 += EXP2F(fmaf(b.z, k2, c2));
    s1 += EXP2F(fmaf(b.w, k1, c1)); s2 += EXP2F(fmaf(b.w, k2, c2));

    s1 = wave_red_sum(s1);
    s2 = wave_red_sum(s2);

    float res;
    if (m > 0.0f)      res = m * s2 * RCPF(s1);
    else if (m < 0.0f) res = m;   // softmax weights sum to 1 -> exactly rho_max
    else               res = 0.0f;

    if (lane == 0) out[cur] = res;
  }
}

extern "C" void kernel_launch(void* const* d_in, const int* in_sizes, int n_in,
                              void* d_out, int out_size, void* d_ws,
                              size_t ws_size, hipStream_t stream) {
  (void)n_in; (void)d_ws; (void)ws_size; (void)out_size;
  const float* x = (const float*)d_in[0];
  const float* scale = (const float*)d_in[1];
  float* out = (float*)d_out;

  const unsigned nrows = (unsigned)(in_sizes[0] / ROW_N); // 524288
  const unsigned rows_per_block = WAVES * ITERS;          // 128
  const unsigned grid = (nrows + rows_per_block - 1) / rows_per_block; // 4096

  hipLaunchKernelGGL(Maxish_25460566130920_kernel, dim3(grid), dim3(256), 0,
                     stream, x, scale, out, nrows);
}